// ssd_multibox_loss_54468775248095
// MI455X (gfx1250) — compile-verified
//
#include <hip/hip_runtime.h>
#include <hip/hip_bf16.h>

typedef float v2f __attribute__((ext_vector_type(2)));
typedef float v8f __attribute__((ext_vector_type(8)));

constexpr int B  = 32;
constexpr int P  = 32768;
constexpr int G  = 64;
constexpr int C  = 21;
constexpr int BG_CLASS = 20;
constexpr int POS_CAP  = 64;   // MAX_NUM/(NEG_RATIO+1)
constexpr int NEG_CAP  = 192;  // MAX_NUM - POS_CAP

// ---------------------------------------------------------------------------
// Exact f32 wave32 sum reduction through the matrix unit:
//   A (16x4 f32): VGPR0 = per-lane value, VGPR1 = 0
//   B (4x16 f32): all ones  =>  D[m,n] = v[m] + v[m+16]
// lane-local sum of the 8 D regs + xor-16 shuffle = full 32-lane sum.
// Requires EXEC == all ones: call only from uniform control flow.
// ---------------------------------------------------------------------------
__device__ __forceinline__ float wave_sum_wmma(float v) {
    v2f a;  a[0] = v;     a[1] = 0.0f;
    v2f bo; bo[0] = 1.0f; bo[1] = 1.0f;
    v8f c = {};
    v8f d = __builtin_amdgcn_wmma_f32_16x16x4_f32(
        /*neg_a=*/false, a, /*neg_b=*/false, bo,
        /*c_mod=*/(short)0, c, /*reuse_a=*/false, /*reuse_b=*/false);
    float s = d[0] + d[1] + d[2] + d[3] + d[4] + d[5] + d[6] + d[7];
    s += __shfl_xor(s, 16, 32);
    return s;
}

// ---------------------------------------------------------------------------
// Kernel 1: per (b,p) best IOU over gts (fits) + argmax gt (idx).
// gts staged in LDS (64 boxes * 4 floats = exactly 256 loads).
// ---------------------------------------------------------------------------
__global__ __launch_bounds__(256)
void k_fits(const float* __restrict__ priors, const float* __restrict__ gt_box,
            float* __restrict__ fits, int* __restrict__ idx) {
    const int b = blockIdx.y;
    const int p = blockIdx.x * 256 + threadIdx.x;

    __shared__ float sg[G * 4];
    sg[threadIdx.x] = gt_box[(size_t)b * G * 4 + threadIdx.x];
    __syncthreads();

    const float4 pc = ((const float4*)priors)[p];      // cxcywh
    const float px1 = pc.x - pc.z * 0.5f, py1 = pc.y - pc.w * 0.5f;
    const float px2 = pc.x + pc.z * 0.5f, py2 = pc.y + pc.w * 0.5f;
    const float aa  = (px2 - px1) * (py2 - py1);

    float best = -1.0f; int bi = 0;
#pragma unroll 4
    for (int g = 0; g < G; ++g) {
        const float gx1 = sg[g*4+0], gy1 = sg[g*4+1];
        const float gx2 = sg[g*4+2], gy2 = sg[g*4+3];
        const float lx = fmaxf(px1, gx1), ly = fmaxf(py1, gy1);
        const float rx = fminf(px2, gx2), ry = fminf(py2, gy2);
        const float w  = fmaxf(rx - lx, 0.0f), h = fmaxf(ry - ly, 0.0f);
        const float inter = w * h;
        const float ab    = (gx2 - gx1) * (gy2 - gy1);
        const float iou   = inter / (aa + ab - inter);
        if (iou > best) { best = iou; bi = g; }      // first-max tie-break
    }
    fits[(size_t)b * P + p] = best;
    idx [(size_t)b * P + p] = bi;
}

// ---------------------------------------------------------------------------
// Kernel 2: per (b,g) argmax prior over all P (recompute IOU, no atomics).
// Packed key (iou_bits<<32)|(P-1-p): max key == max iou, ties -> lowest p.
// ---------------------------------------------------------------------------
__global__ __launch_bounds__(256)
void k_bestp(const float* __restrict__ priors, const float* __restrict__ gt_box,
             int* __restrict__ bestp) {
    const int b = blockIdx.y, g = blockIdx.x;
    const float4 gb = ((const float4*)gt_box)[b * G + g];  // corner form
    const float ab  = (gb.z - gb.x) * (gb.w - gb.y);

    unsigned long long key = 0ull;
    for (int p = threadIdx.x; p < P; p += 256) {
        const float4 pc = ((const float4*)priors)[p];
        const float px1 = pc.x - pc.z * 0.5f, py1 = pc.y - pc.w * 0.5f;
        const float px2 = pc.x + pc.z * 0.5f, py2 = pc.y + pc.w * 0.5f;
        const float aa  = (px2 - px1) * (py2 - py1);
        const float lx = fmaxf(px1, gb.x), ly = fmaxf(py1, gb.y);
        const float rx = fminf(px2, gb.z), ry = fminf(py2, gb.w);
        const float w  = fmaxf(rx - lx, 0.0f), h = fmaxf(ry - ly, 0.0f);
        const float inter = w * h;
        const float iou   = inter / (aa + ab - inter);   // >= 0
        const unsigned long long k =
            ((unsigned long long)__float_as_uint(iou) << 32) |
            (unsigned int)(P - 1 - p);
        key = (k > key) ? k : key;
    }
    __shared__ unsigned long long sk[256];
    sk[threadIdx.x] = key;
    __syncthreads();
    for (int s = 128; s > 0; s >>= 1) {
        if (threadIdx.x < s) {
            const unsigned long long o = sk[threadIdx.x + s];
            if (o > sk[threadIdx.x]) sk[threadIdx.x] = o;
        }
        __syncthreads();
    }
    if (threadIdx.x == 0)
        bestp[b * G + g] = P - 1 - (int)(sk[0] & 0xffffffffu);
}

// ---------------------------------------------------------------------------
// Kernel 3: force-match scatter. One thread per batch, sequential over g so
// later g wins collisions (matches JAX .at[].set ordering).
// ---------------------------------------------------------------------------
__global__ __launch_bounds__(32)
void k_force(const int* __restrict__ bestp, float* __restrict__ fits,
             int* __restrict__ idx) {
    const int b = threadIdx.x;
    for (int g = 0; g < G; ++g) {
        const int p = bestp[b * G + g];
        fits[(size_t)b * P + p] = 2.0f;
        idx [(size_t)b * P + p] = g;
    }
}

// ---------------------------------------------------------------------------
// Kernel 4: per-batch hard-negative mining. Pass A: counts -> eff_pos/neg_num.
// Pass B: ballot/popcount block scan, emit compacted selected-prior list:
//   list[b][rank]      = p  for selected positives (rank < eff_pos <= 64)
//   list[b][64 + rank] = p  for selected negatives (rank < neg_num <= 192)
// ---------------------------------------------------------------------------
__global__ __launch_bounds__(1024)
void k_scan(const float* __restrict__ fits, int* __restrict__ list,
            float* __restrict__ cnt_f, int* __restrict__ cnt_i) {
    const int b    = blockIdx.x;
    const int wid  = threadIdx.x >> 5;
    const int lane = threadIdx.x & 31;
    const float* f = fits + (size_t)b * P;

    // ---- Pass A: totals ----
    int posc = 0, negc = 0;
    for (int i = threadIdx.x; i < P; i += 1024) {
        const float v = f[i];
        posc += (v > 0.5f);
        negc += (v < 0.5f);
    }
    for (int o = 16; o > 0; o >>= 1) {
        posc += __shfl_xor(posc, o, 32);
        negc += __shfl_xor(negc, o, 32);
    }
    __shared__ int sp[32], sn[32];
    __shared__ int s_eff, s_negnum;
    if (lane == 0) { sp[wid] = posc; sn[wid] = negc; }
    __syncthreads();
    if (threadIdx.x == 0) {
        int pc = 0, nc = 0;
        for (int w = 0; w < 32; ++w) { pc += sp[w]; nc += sn[w]; }
        const int eff    = pc < POS_CAP ? pc : POS_CAP;
        const int negnum = (pc <= POS_CAP) ? (int)((float)pc * 3.0f) : NEG_CAP;
        const int negsel = nc < negnum ? nc : negnum;
        s_eff = eff; s_negnum = negnum;
        cnt_f[2 * b + 0] = fmaxf((float)eff,    1.0f);   // n_pos
        cnt_f[2 * b + 1] = fmaxf((float)negsel, 1.0f);   // n_neg
        cnt_i[2 * b + 0] = eff;
        cnt_i[2 * b + 1] = negsel;
    }
    __syncthreads();
    const int eff = s_eff, negnum = s_negnum;

    // ---- Pass B: scan + compaction ----
    __shared__ int wpT[32], wnT[32], wpP[33], wnP[33];
    int posCarry = 0, negCarry = 0;
    const unsigned int ltm = (1u << lane) - 1u;
    for (int base = 0; base < P; base += 1024) {
        const int i   = base + threadIdx.x;
        const float v = f[i];
        const bool pos = (v > 0.5f), neg = (v < 0.5f);
        const unsigned int pb = (unsigned int)__ballot(pos);
        const unsigned int nb = (unsigned int)__ballot(neg);
        const int posLane = __popc(pb & ltm);
        const int negLane = __popc(nb & ltm);
        if (lane == 0) { wpT[wid] = __popc(pb); wnT[wid] = __popc(nb); }
        __syncthreads();
        if (threadIdx.x == 0) {
            int a = 0, c2 = 0;
            for (int w = 0; w < 32; ++w) {
                wpP[w] = a;  a  += wpT[w];
                wnP[w] = c2; c2 += wnT[w];
            }
            wpP[32] = a; wnP[32] = c2;
        }
        __syncthreads();
        const int pr = posCarry + wpP[wid] + posLane;    // exclusive rank
        const int nr = negCarry + wnP[wid] + negLane;
        if (pos && pr < eff)    list[b * 256 + pr]      = i;
        if (neg && nr < negnum) list[b * 256 + 64 + nr] = i;
        posCarry += wpP[32];
        negCarry += wnP[32];
        if (posCarry >= eff && negCarry >= negnum) break;   // uniform
        __syncthreads();
    }
}

// ---------------------------------------------------------------------------
// Kernel 5: loss on selected priors only. One 256-thread block per batch:
// threads 0..63 handle positives, 64..255 negatives. Block reduce via WMMA
// wave reduction (EXEC all ones; called from uniform control flow).
// ---------------------------------------------------------------------------
__global__ __launch_bounds__(256)
void k_loss(const float* __restrict__ pred_cls, const float* __restrict__ pred_reg,
            const int* __restrict__ gt_cls, const float* __restrict__ gt_box,
            const float* __restrict__ priors, const int* __restrict__ idx,
            const int* __restrict__ list, const int* __restrict__ cnt_i,
            float* __restrict__ sums) {
    const int b = blockIdx.x, t = threadIdx.x;
    const int eff  = cnt_i[2 * b + 0];
    const int nneg = cnt_i[2 * b + 1];
    const bool isPos = (t < 64)  && (t < eff);
    const bool isNeg = (t >= 64) && (t - 64 < nneg);

    float posv = 0.0f, regv = 0.0f, negv = 0.0f;
    if (isPos || isNeg) {
        const int p = list[b * 256 + t];
        const float* lg = pred_cls + ((size_t)b * P + p) * C;
        float lv[C];
        float m = -3.4e38f;
#pragma unroll
        for (int c = 0; c < C; ++c) { lv[c] = lg[c]; m = fmaxf(m, lv[c]); }
        float s = 0.0f;
#pragma unroll
        for (int c = 0; c < C; ++c) s += expf(lv[c] - m);
        const float lse = m + logf(s);

        if (isPos) {
            const int g  = idx[(size_t)b * P + p];
            const int tc = gt_cls[b * G + g];
            posv = lse - lv[tc];
            const float4 gb = ((const float4*)gt_box)[b * G + g];   // corners
            const float gcx = (gb.x + gb.z) * 0.5f, gcy = (gb.y + gb.w) * 0.5f;
            const float gw  = gb.z - gb.x,          gh  = gb.w - gb.y;
            const float4 pc = ((const float4*)priors)[p];           // cxcywh
            const float tx = (gcx - pc.x) / pc.z;
            const float ty = (gcy - pc.y) / pc.w;
            const float tw = logf(gw / pc.z);
            const float th = logf(gh / pc.w);
            const float4 pr = ((const float4*)pred_reg)[(size_t)b * P + p];
            regv = fabsf(pr.x - tx) + fabsf(pr.y - ty) +
                   fabsf(pr.z - tw) + fabsf(pr.w - th);
        } else {
            negv = lse - lv[BG_CLASS];
        }
    }

    // uniform: every thread participates (zeros for inactive)
    const float rp = wave_sum_wmma(posv);
    const float rr = wave_sum_wmma(regv);
    const float rn = wave_sum_wmma(negv);

    __shared__ float sb[8][3];
    const int wid = t >> 5, lane = t & 31;
    if (lane == 0) { sb[wid][0] = rp; sb[wid][1] = rr; sb[wid][2] = rn; }
    __syncthreads();
    if (t == 0) {
        float a = 0.0f, c2 = 0.0f, d = 0.0f;
        for (int w = 0; w < 8; ++w) { a += sb[w][0]; c2 += sb[w][1]; d += sb[w][2]; }
        sums[b * 3 + 0] = a;
        sums[b * 3 + 1] = c2;
        sums[b * 3 + 2] = d;
    }
}

// ---------------------------------------------------------------------------
// Kernel 6: final mean over batches. One wave; lane = batch.
// ---------------------------------------------------------------------------
__global__ __launch_bounds__(32)
void k_final(const float* __restrict__ sums, const float* __restrict__ cnt_f,
             float* __restrict__ out) {
    const int b = threadIdx.x;
    const float np = cnt_f[2 * b + 0];
    const float nn = cnt_f[2 * b + 1];
    const float lb = sums[b * 3 + 0] / np +
                     sums[b * 3 + 1] / (np * 4.0f) +
                     sums[b * 3 + 2] / nn;
    const float tot = wave_sum_wmma(lb);
    if (b == 0) out[0] = tot * (1.0f / (float)B);
}

// ---------------------------------------------------------------------------
extern "C" void kernel_launch(void* const* d_in, const int* in_sizes, int n_in,
                              void* d_out, int out_size, void* d_ws, size_t ws_size,
                              hipStream_t stream) {
    const float* pred_cls = (const float*)d_in[0];   // (B,P,C)
    const float* pred_reg = (const float*)d_in[1];   // (B,P,4)
    const int*   gt_cls   = (const int*)  d_in[2];   // (B,G)
    const float* gt_box   = (const float*)d_in[3];   // (B,G,4) corners
    const float* priors   = (const float*)d_in[4];   // (P,4) cxcywh
    float* out = (float*)d_out;

    // workspace layout (all 4-byte types; base is >=256B aligned)
    char* ws = (char*)d_ws;
    float* fits  = (float*)ws;                      ws += (size_t)B * P * sizeof(float);
    int*   idx   = (int*)ws;                        ws += (size_t)B * P * sizeof(int);
    int*   bestp = (int*)ws;                        ws += (size_t)B * G * sizeof(int);
    int*   list  = (int*)ws;                        ws += (size_t)B * 256 * sizeof(int);
    float* cnt_f = (float*)ws;                      ws += (size_t)B * 2 * sizeof(float);
    int*   cnt_i = (int*)ws;                        ws += (size_t)B * 2 * sizeof(int);
    float* sums  = (float*)ws;                      ws += (size_t)B * 3 * sizeof(float);
    (void)in_sizes; (void)n_in; (void)out_size; (void)ws_size;

    k_fits <<<dim3(P / 256, B), 256, 0, stream>>>(priors, gt_box, fits, idx);
    k_bestp<<<dim3(G, B),       256, 0, stream>>>(priors, gt_box, bestp);
    k_force<<<1, 32, 0, stream>>>(bestp, fits, idx);
    k_scan <<<B, 1024, 0, stream>>>(fits, list, cnt_f, cnt_i);
    k_loss <<<B, 256, 0, stream>>>(pred_cls, pred_reg, gt_cls, gt_box, priors,
                                   idx, list, cnt_i, sums);
    k_final<<<1, 32, 0, stream>>>(sums, cnt_f, out);
}